// KnowledgeRetriever_85830626443682
// MI455X (gfx1250) — compile-verified
//
#include <hip/hip_runtime.h>
#include <hip/hip_bf16.h>
#include <math.h>

typedef __attribute__((ext_vector_type(2))) float v2f;
typedef __attribute__((ext_vector_type(4))) float v4f;
typedef __attribute__((ext_vector_type(8))) float v8f;

#define WMMA_F32(A, B, C) \
    __builtin_amdgcn_wmma_f32_16x16x4_f32(false, (A), false, (B), (short)0, (C), false, false)

// ---------------------------------------------------------------------------
// Generic C = A(M x Kd) * B(N x Kd)^T (+ bias, optional relu), all fp32.
// V_WMMA_F32_16X16X4_F32; each wave owns a 16x64 tile (4 accums, A reused 4x).
// Two-stage software pipeline: while the WMMA group for step k executes, the
// loads for steps k+1 AND k+2 are in flight, so waits are partial
// (loadcnt <= 5..8) instead of stalling on the just-issued load group.
// Requires M % 16 == 0, N % 64 == 0, Kd % 4 == 0, Kd >= 8 (true here).
// ---------------------------------------------------------------------------
__global__ __launch_bounds__(256) void gemm_nt_f32_wmma(
    const float* __restrict__ A, int lda,
    const float* __restrict__ Bm, int ldb,
    const float* __restrict__ bias,
    float* __restrict__ C, int ldc,
    int M, int Nout, int Kdim, int do_relu)
{
    const int lane = threadIdx.x & 31;
    const int wave = threadIdx.x >> 5;
    const int m0 = blockIdx.y << 4;
    const int n0 = blockIdx.x * (8 * 64) + wave * 64;
    if (n0 >= Nout || m0 >= M) return;           // wave-uniform: EXEC stays all-1s

    const int half = lane >> 4;                  // selects K pair {0,1} vs {2,3}
    const int r    = lane & 15;                  // M index (A) / N index (B)

    const float* Arow = A  + (size_t)(m0 + r) * lda + 2 * half;
    const float* Brow = Bm + (size_t)(n0 + r) * ldb + 2 * half;
    const size_t bstep = (size_t)16 * ldb;

    v8f acc0 = {}, acc1 = {}, acc2 = {}, acc3 = {};

    // Stage 0 fragments (k = 0)
    v2f a0f = *(const v2f*)(Arow);
    v2f b00 = *(const v2f*)(Brow);
    v2f b01 = *(const v2f*)(Brow + bstep);
    v2f b02 = *(const v2f*)(Brow + bstep * 2);
    v2f b03 = *(const v2f*)(Brow + bstep * 3);
    // Stage 1 fragments (k = 4)
    v2f a1f = *(const v2f*)(Arow + 4);
    v2f b10 = *(const v2f*)(Brow + 4);
    v2f b11 = *(const v2f*)(Brow + bstep     + 4);
    v2f b12 = *(const v2f*)(Brow + bstep * 2 + 4);
    v2f b13 = *(const v2f*)(Brow + bstep * 3 + 4);

#pragma unroll 2
    for (int k0 = 8; k0 < Kdim; k0 += 4) {
        // Issue loads for step k0 (two steps ahead of the WMMA group below).
        v2f an  = *(const v2f*)(Arow + k0);
        v2f bn0 = *(const v2f*)(Brow + k0);
        v2f bn1 = *(const v2f*)(Brow + bstep     + k0);
        v2f bn2 = *(const v2f*)(Brow + bstep * 2 + k0);
        v2f bn3 = *(const v2f*)(Brow + bstep * 3 + k0);
        // Consume the oldest stage.
        acc0 = WMMA_F32(a0f, b00, acc0);
        acc1 = WMMA_F32(a0f, b01, acc1);
        acc2 = WMMA_F32(a0f, b02, acc2);
        acc3 = WMMA_F32(a0f, b03, acc3);
        // Rotate pipeline.
        a0f = a1f; b00 = b10; b01 = b11; b02 = b12; b03 = b13;
        a1f = an;  b10 = bn0; b11 = bn1; b12 = bn2; b13 = bn3;
    }
    // Drain both stages.
    acc0 = WMMA_F32(a0f, b00, acc0);
    acc1 = WMMA_F32(a0f, b01, acc1);
    acc2 = WMMA_F32(a0f, b02, acc2);
    acc3 = WMMA_F32(a0f, b03, acc3);
    acc0 = WMMA_F32(a1f, b10, acc0);
    acc1 = WMMA_F32(a1f, b11, acc1);
    acc2 = WMMA_F32(a1f, b12, acc2);
    acc3 = WMMA_F32(a1f, b13, acc3);

    const int rowbase = m0 + 8 * half;           // C/D layout: M split by 8 across halves
#define EPILOGUE(ACC, T)                                                     \
    do {                                                                     \
        int col = n0 + (T) * 16 + r;                                         \
        float bv = bias ? bias[col] : 0.0f;                                  \
        float* Cp = C + (size_t)rowbase * ldc + col;                         \
        _Pragma("unroll")                                                    \
        for (int v = 0; v < 8; ++v) {                                        \
            float x = ACC[v] + bv;                                           \
            if (do_relu) x = fmaxf(x, 0.0f);                                 \
            Cp[(size_t)v * ldc] = x;                                         \
        }                                                                    \
    } while (0)
    EPILOGUE(acc0, 0);
    EPILOGUE(acc1, 1);
    EPILOGUE(acc2, 2);
    EPILOGUE(acc3, 3);
#undef EPILOGUE
}

// ---------------------------------------------------------------------------
// scores[b,n] = sum_j relu(qf[b,j] + kp[n,j]) * S2[j] + bs2
// One wave32 per (b,n); consecutive waves share the kp row (L2 reuse).
// float4 lane-strided loads -> global_load_b128, 4 iterations for K=512.
// ---------------------------------------------------------------------------
__global__ __launch_bounds__(256) void scores_kernel(
    const float* __restrict__ qf, const float* __restrict__ kp,
    const float* __restrict__ S2, const float* __restrict__ bs2,
    float* __restrict__ scores, int Bn, int Nn, int Kn)
{
    const int wid  = (int)((blockIdx.x * blockDim.x + threadIdx.x) >> 5);
    const int lane = threadIdx.x & 31;
    if (wid >= Bn * Nn) return;
    const int b = wid % Bn;
    const int n = wid / Bn;

    const float* qrow = qf + (size_t)b * Kn;
    const float* krow = kp + (size_t)n * Kn;

    float s = 0.0f;
    for (int j = lane * 4; j < Kn; j += 128) {
        v4f q = *(const v4f*)(qrow + j);
        v4f k = *(const v4f*)(krow + j);
        v4f w = *(const v4f*)(S2 + j);
#pragma unroll
        for (int c = 0; c < 4; ++c) {
            float v = fmaxf(q[c] + k[c], 0.0f);
            s = fmaf(v, w[c], s);
        }
    }
#pragma unroll
    for (int off = 16; off > 0; off >>= 1)
        s += __shfl_xor(s, off, 32);
    if (lane == 0) scores[(size_t)b * Nn + n] = s + bs2[0];
}

// ---------------------------------------------------------------------------
// Per batch row: 32x iterative argmax (tie-break lower index, like lax.top_k),
// then gather the selected knowledge rows into the output (b128 copies).
// ---------------------------------------------------------------------------
__global__ __launch_bounds__(256) void topk_gather_kernel(
    float* __restrict__ scores, const float* __restrict__ knowledge,
    float* __restrict__ out_retrieved, float* __restrict__ out_scores,
    int Nn, int Kn, int TopK)
{
    const int b = blockIdx.x;
    const int t = threadIdx.x;
    float* row = scores + (size_t)b * Nn;

    __shared__ float sval[256];
    __shared__ int   sidx[256];
    __shared__ int   topIdx[64];

    for (int iter = 0; iter < TopK; ++iter) {
        float best = -INFINITY;
        int   bidx = 0x7fffffff;
        for (int n = t; n < Nn; n += 256) {
            float v = row[n];
            if (v > best) { best = v; bidx = n; }   // increasing n => first hit wins ties
        }
        sval[t] = best; sidx[t] = bidx;
        __syncthreads();
        for (int s = 128; s > 0; s >>= 1) {
            if (t < s) {
                float v2 = sval[t + s]; int i2 = sidx[t + s];
                if (v2 > sval[t] || (v2 == sval[t] && i2 < sidx[t])) {
                    sval[t] = v2; sidx[t] = i2;
                }
            }
            __syncthreads();
        }
        if (t == 0) {
            topIdx[iter] = sidx[0];
            out_scores[b * TopK + iter] = sval[0];
            row[sidx[0]] = -INFINITY;               // recomputed by scores_kernel each call
        }
        __syncthreads();
    }

    const int kv = Kn >> 2;                          // float4 elements per row
    for (int i = 0; i < TopK; ++i) {
        const int idx = topIdx[i];
        const v4f* src = (const v4f*)(knowledge + (size_t)idx * Kn);
        v4f* dst = (v4f*)(out_retrieved + ((size_t)(b * TopK + i)) * Kn);
        for (int j = t; j < kv; j += 256) dst[j] = src[j];
    }
}

extern "C" void kernel_launch(void* const* d_in, const int* in_sizes, int n_in,
                              void* d_out, int out_size, void* d_ws, size_t ws_size,
                              hipStream_t stream) {
    (void)in_sizes; (void)n_in; (void)out_size; (void)ws_size;

    const float* ctx       = (const float*)d_in[0];   // (16, 1024)
    const float* knowledge = (const float*)d_in[1];   // (8000, 512)
    const float* W1        = (const float*)d_in[2];   // (1024, 1024)
    const float* b1        = (const float*)d_in[3];   // (1024,)
    const float* W2        = (const float*)d_in[4];   // (512, 1024)
    const float* b2        = (const float*)d_in[5];   // (512,)
    const float* S1        = (const float*)d_in[6];   // (512, 1024)
    const float* bs1       = (const float*)d_in[7];   // (512,)
    const float* S2        = (const float*)d_in[8];   // (1, 512)
    const float* bs2       = (const float*)d_in[9];   // (1,)
    // d_in[10] = top_k scalar; fixed at 32 per the reference setup

    const int Bq = 16, Nk = 8000, D = 1024, K = 512, TopK = 32;

    float* ws   = (float*)d_ws;
    float* d_h  = ws;                         // 16*1024
    float* d_q  = d_h  + (size_t)Bq * D;      // 16*512
    float* d_qf = d_q  + (size_t)Bq * K;      // 16*512
    float* d_kp = d_qf + (size_t)Bq * K;      // 8000*512
    float* d_sc = d_kp + (size_t)Nk * K;      // 16*8000

    // 1. h = relu(ctx @ W1^T + b1)           M=16, N=1024, K=1024
    gemm_nt_f32_wmma<<<dim3(D / 512, Bq / 16), 256, 0, stream>>>(
        ctx, D, W1, D, b1, d_h, D, Bq, D, D, 1);
    // 2. q = h @ W2^T + b2                   M=16, N=512, K=1024
    gemm_nt_f32_wmma<<<dim3(1, Bq / 16), 256, 0, stream>>>(
        d_h, D, W2, D, b2, d_q, K, Bq, K, D, 0);
    // 3. qf = q @ S1q^T + bs1  (S1[:, :K])   M=16, N=512, K=512
    gemm_nt_f32_wmma<<<dim3(1, Bq / 16), 256, 0, stream>>>(
        d_q, K, S1, 2 * K, bs1, d_qf, K, Bq, K, K, 0);
    // 4. kp = knowledge @ S1k^T (S1[:, K:])  M=8000, N=512, K=512  (the big one)
    gemm_nt_f32_wmma<<<dim3(1, Nk / 16), 256, 0, stream>>>(
        knowledge, K, S1 + K, 2 * K, nullptr, d_kp, K, Nk, K, K, 0);
    // 5. fused relu-add-dot scores
    {
        const long long totalThreads = (long long)Bq * Nk * 32;
        const int blocks = (int)((totalThreads + 255) / 256);
        scores_kernel<<<blocks, 256, 0, stream>>>(d_qf, d_kp, S2, bs2, d_sc, Bq, Nk, K);
    }
    // 6. top-k + gather into d_out: [retrieved (16*32*512) | top_scores (16*32)]
    float* out = (float*)d_out;
    topk_gather_kernel<<<Bq, 256, 0, stream>>>(
        d_sc, knowledge, out, out + (size_t)Bq * TopK * K, Nk, K, TopK);
}